// HyboMessagePassing_66992899883289
// MI455X (gfx1250) — compile-verified
//
#include <hip/hip_runtime.h>
#include <hip/hip_bf16.h>
#include <stdint.h>

// 16-byte vector types for b128 transfers
typedef int   v4i __attribute__((vector_size(16)));
typedef float v4f __attribute__((vector_size(16)));
typedef __attribute__((address_space(1))) v4i* gv4i_p;
typedef __attribute__((address_space(3))) v4i* lv4i_p;

#if defined(__HIP_DEVICE_COMPILE__) && defined(__gfx1250__) && \
    __has_builtin(__builtin_amdgcn_global_load_async_to_lds_b128)
#define USE_ASYNC_LDS 1
#else
#define USE_ASYNC_LDS 0
#endif

template <int N>
__device__ __forceinline__ void wait_async_le() {
#if defined(__HIP_DEVICE_COMPILE__) && defined(__gfx1250__)
#if __has_builtin(__builtin_amdgcn_s_wait_asynccnt)
  __builtin_amdgcn_s_wait_asynccnt(N);
#else
  asm volatile("s_wait_asynccnt %0" ::"i"(N) : "memory");
#endif
#endif
}

__device__ __forceinline__ void wait_ds_zero() {
#if defined(__HIP_DEVICE_COMPILE__) && defined(__gfx1250__)
  asm volatile("s_wait_dscnt 0" ::: "memory");
#endif
}

// Explicit addrspace(3) load so the compiler must emit ds_load_b128.
__device__ __forceinline__ v4f lds_load_f4(const float* p) {
#if defined(__HIP_DEVICE_COMPILE__) && defined(__gfx1250__)
  return *(const __attribute__((address_space(3))) v4f*)p;
#else
  return *(const v4f*)p;
#endif
}

#define DFIX 128  // feature width specialization (reference uses D=128)

// -------- Main kernel (emitted first so the disasm snippet shows it) --------
// One wave per edge. Double-buffered async global->LDS staging of the streaming
// edge row overlaps with the L2-resident gather of the source-node row.
// Each lane owns exactly one float4 of the 128-wide feature vector.
__global__ void __launch_bounds__(256)
hybo_scatter_d128(const float* __restrict__ input, const float* __restrict__ edge_input,
                  const float* __restrict__ ew, const int* __restrict__ node_in,
                  const int* __restrict__ node_out, const int* __restrict__ rel,
                  const int* __restrict__ counts, float* __restrict__ out,
                  int E, const int* __restrict__ Rp) {
  __shared__ float smem[8 * 2 * DFIX];           // 8 waves x 2 buffers x 512B = 8KB
  const int R = *Rp;
  const float invR = 1.0f / (float)R;
  const int lane = (int)(threadIdx.x & 31);
  const int wave = (int)(threadIdx.x >> 5);
  float* buf[2];
  buf[0] = smem + wave * 2 * DFIX;
  buf[1] = buf[0] + DFIX;

  const long long wid = (long long)blockIdx.x * 8 + wave;   // wave-uniform
  const long long nw  = (long long)gridDim.x * 8;

  long long e = wid;
  if (e >= (long long)E) return;                 // whole wave exits together

  int cur = 0;
#if USE_ASYNC_LDS
  // Prologue: stage first edge row (one async instruction per wave, 16B/lane).
  __builtin_amdgcn_global_load_async_to_lds_b128(
      (gv4i_p)(edge_input + (size_t)e * DFIX + lane * 4),
      (lv4i_p)(buf[0] + lane * 4), 0, 0);
#endif

  for (; e < (long long)E; e += nw) {
    const long long en = e + nw;
    const int ni = node_in[e];                   // wave-uniform -> scalar loads
    const int no = node_out[e];
    const int rr = rel[e];
    const int c  = counts[no * R + rr];
    const float scale = ew[e] * invR / (float)(c < 1 ? 1 : c);

    if (en < (long long)E) {
#if USE_ASYNC_LDS
      wait_ds_zero();                            // prior reads of buf[cur^1] retired
      __builtin_amdgcn_global_load_async_to_lds_b128(
          (gv4i_p)(edge_input + (size_t)en * DFIX + lane * 4),
          (lv4i_p)(buf[cur ^ 1] + lane * 4), 0, 0);
#endif
      // warm next gathered node row while copies fly
      __builtin_prefetch(input + (size_t)node_in[en] * DFIX + lane * 4, 0, 1);
#if USE_ASYNC_LDS
      wait_async_le<1>();                        // oldest copy (buf[cur]) done
#endif
    } else {
#if USE_ASYNC_LDS
      wait_async_le<0>();
#endif
    }

    const v4f a = *(const v4f*)(input + (size_t)ni * DFIX + lane * 4);  // L2 gather
#if USE_ASYNC_LDS
    const v4f b = lds_load_f4(buf[cur] + lane * 4);                     // ds_load_b128
#else
    const v4f b = *(const v4f*)(edge_input + (size_t)e * DFIX + lane * 4);
#endif
    float* orow = out + (size_t)no * DFIX + lane * 4;
    atomicAdd(orow + 0, (a[0] + b[0]) * scale);
    atomicAdd(orow + 1, (a[1] + b[1]) * scale);
    atomicAdd(orow + 2, (a[2] + b[2]) * scale);
    atomicAdd(orow + 3, (a[3] + b[3]) * scale);
    cur ^= 1;
  }
}

// -------- Generic fallback for D != 128 (plain global path) --------
__global__ void hybo_scatter_generic(const float* __restrict__ input,
                                     const float* __restrict__ edge_input,
                                     const float* __restrict__ ew,
                                     const int* __restrict__ node_in,
                                     const int* __restrict__ node_out,
                                     const int* __restrict__ rel,
                                     const int* __restrict__ counts,
                                     float* __restrict__ out,
                                     int E, int D, const int* __restrict__ Rp) {
  const int R = *Rp;
  const float invR = 1.0f / (float)R;
  const long long total = (long long)E * D;
  const long long stride = (long long)gridDim.x * blockDim.x;
  for (long long i = (long long)blockIdx.x * blockDim.x + threadIdx.x; i < total; i += stride) {
    const long long e = i / D;
    const int d = (int)(i - e * D);
    const int no = node_out[e];
    const int c  = counts[no * R + rel[e]];
    const float scale = ew[e] * invR / (float)(c < 1 ? 1 : c);
    const float v = (input[(size_t)node_in[e] * D + d] + edge_input[(size_t)e * D + d]) * scale;
    atomicAdd(out + (size_t)no * D + d, v);
  }
}

// -------- Zero output + per-(node,relation) counters --------
__global__ void hybo_init(float* __restrict__ out, int* __restrict__ counts,
                          long long nd, long long n, const int* __restrict__ Rp) {
  const long long R = *Rp;
  const long long total = nd + n * R;
  const long long stride = (long long)gridDim.x * blockDim.x;
  for (long long i = (long long)blockIdx.x * blockDim.x + threadIdx.x; i < total; i += stride) {
    if (i < nd) out[i] = 0.0f;
    else        counts[i - nd] = 0;
  }
}

// -------- Bucket counts (1.4 MB, L2-resident u32 atomics) --------
__global__ void hybo_count(const int* __restrict__ node_out, const int* __restrict__ rel,
                           int* __restrict__ counts, int E, const int* __restrict__ Rp) {
  const int R = *Rp;
  const int stride = gridDim.x * blockDim.x;
  for (int e = blockIdx.x * blockDim.x + threadIdx.x; e < E; e += stride) {
    atomicAdd(&counts[node_out[e] * R + rel[e]], 1);
  }
}

extern "C" void kernel_launch(void* const* d_in, const int* in_sizes, int n_in,
                              void* d_out, int out_size, void* d_ws, size_t ws_size,
                              hipStream_t stream) {
  const float* input      = (const float*)d_in[0];   // [N, D]
  const float* edge_input = (const float*)d_in[1];   // [E, D]
  const float* edge_w     = (const float*)d_in[2];   // [E]
  const int*   node_in    = (const int*)d_in[3];     // [E]
  const int*   node_out   = (const int*)d_in[4];     // [E]
  const int*   relation   = (const int*)d_in[5];     // [E]
  const int*   Rp         = (const int*)d_in[7];     // scalar num_relation (device)
  float* out  = (float*)d_out;
  int* counts = (int*)d_ws;                          // N*R ints of scratch

  const int E  = in_sizes[2];
  const int D  = (E > 0) ? in_sizes[1] / E : 1;
  const long long ND = in_sizes[0];
  const long long N  = ND / D;

  hybo_init<<<1024, 256, 0, stream>>>(out, counts, ND, N, Rp);

  int cblocks = (E + 255) / 256; if (cblocks > 4096) cblocks = 4096; if (cblocks < 1) cblocks = 1;
  hybo_count<<<cblocks, 256, 0, stream>>>(node_out, relation, counts, E, Rp);

  if (D == DFIX) {
    hybo_scatter_d128<<<2048, 256, 0, stream>>>(input, edge_input, edge_w, node_in,
                                                node_out, relation, counts, out, E, Rp);
  } else {
    hybo_scatter_generic<<<2048, 256, 0, stream>>>(input, edge_input, edge_w, node_in,
                                                   node_out, relation, counts, out, E, D, Rp);
  }
}